// pre_module_49331994362578
// MI455X (gfx1250) — compile-verified
//
#include <hip/hip_runtime.h>

// Problem geometry (reference: x = (8, 4, 1024, 1024) f32, out = (8, 15, 1024, 1024) f32)
#define IMG_W 1024
#define TILE  64
#define HALO  5                    // 5 successive 3x3 blurs -> 5-px row halo
#define XOFF  8                    // left column halo (16B-aligned async chunks)
#define LTH   (TILE + 2 * HALO)    // 74 rows
#define LTW   84                   // 84 cols = 21 float4 chunks (left halo 8, right halo 12)
#define NCH   21                   // float4 chunks per row
#define NLS   (LTH * LTW)          // 6216 LDS cells per buffer
#define NTH   256                  // 8 wave32s

__device__ __forceinline__ int iclamp(int v, int lo, int hi) {
  return v < lo ? lo : (v > hi ? hi : v);
}

__device__ __forceinline__ void wait_async_all() {
#if defined(__has_builtin) && __has_builtin(__builtin_amdgcn_s_wait_asynccnt)
  __builtin_amdgcn_s_wait_asynccnt(0);
#else
  asm volatile("s_wait_asynccnt 0" ::: "memory");
#endif
}

// CDNA5 async global->LDS DMA, 16 bytes per lane (tracked by ASYNCcnt, no VGPR data path).
__device__ __forceinline__ void async_load_lds_b128(const float* gp, const float* lds_cell) {
  // Low 32 bits of a generic pointer to LDS are the LDS byte offset (ISA 10.2 aperture map).
  unsigned loff = (unsigned)(unsigned long long)(const void*)lds_cell;
  asm volatile("global_load_async_to_lds_b128 %0, %1, off"
               :: "v"(loff), "v"(gp)
               : "memory");
}

// Store the exact 64x64 interior of LDS tile A to output channel `ch` as float4s.
// Interior offset (r+5)*84 + 8 is 16B-aligned -> ds_load_b128 + global_store_b128.
__device__ __forceinline__ void store_tile_channel(const float* __restrict__ A,
                                                   float* __restrict__ on, int ch,
                                                   int gy0, int gx0, int tid) {
  for (int i = tid; i < (TILE * TILE) / 4; i += NTH) {
    int r = i >> 4;            // tile row 0..63
    int q = (i & 15) << 2;     // quad start col
    const float* s = &A[(r + HALO) * LTW + XOFF + q];
    float4 v = make_float4(s[0], s[1], s[2], s[3]);
    *reinterpret_cast<float4*>(on + ((size_t)ch * IMG_W + (gy0 + r)) * IMG_W + gx0 + q) = v;
  }
}

__global__ __launch_bounds__(NTH)
void matte_prep_kernel(const float* __restrict__ x, float* __restrict__ out) {
  __shared__ __align__(16) float Tm[NLS];   // raw trimap tile (kept for both chains)
  __shared__ __align__(16) float Ab[NLS];   // current stage value
  __shared__ __align__(16) float Hb[NLS];   // horizontal-pass temp

  const int tid = threadIdx.x;
  const int gx0 = blockIdx.x * TILE;
  const int gy0 = blockIdx.y * TILE;
  const int n   = blockIdx.z;

  const float* __restrict__ xin = x + (size_t)n * 4 * IMG_W * IMG_W;
  const float* __restrict__ tri = xin + (size_t)3 * IMG_W * IMG_W;
  float* __restrict__ on = out + (size_t)n * 15 * IMG_W * IMG_W;

  // ---- Kick off async B128 DMA of the 74x84 trimap tile into LDS. Chunk bases are
  //      4-col aligned and 1024 % 4 == 0, so a chunk is never partially out of a row:
  //      clamping keeps every address inside the trimap plane; clamped (fully-OOB)
  //      chunks land in cells the zero pass below overwrites anyway. ----
  for (int i = tid; i < LTH * NCH; i += NTH) {
    int row = i / NCH;
    int k   = i - row * NCH;
    int gy  = iclamp(gy0 + row - HALO, 0, IMG_W - 1);
    int c0  = iclamp(gx0 - XOFF + 4 * k, 0, IMG_W - 4);
    async_load_lds_b128(tri + (size_t)gy * IMG_W + c0, &Tm[row * LTW + 4 * k]);
  }

  // ---- Overlap: stream RGB channels 0..2 straight through while the DMA is in flight. ----
  for (int i = tid; i < 3 * (TILE * TILE) / 4; i += NTH) {
    int c  = i >> 10;          // channel 0..2
    int rq = i & 1023;
    int r  = rq >> 4;
    int q  = (rq & 15) << 2;
    size_t off = ((size_t)c * IMG_W + (gy0 + r)) * IMG_W + gx0 + q;
    float4 v = *reinterpret_cast<const float4*>(xin + off);
    *reinterpret_cast<float4*>(on + off) = v;
  }

  wait_async_all();
  __syncthreads();

  // ---- Zero the cells outside the image (reproduces 'SAME' zero padding). ----
  for (int i = tid; i < NLS; i += NTH) {
    int ty = i / LTW;
    int tx = i - ty * LTW;
    int gy = gy0 + ty - HALO;
    int gx = gx0 + tx - XOFF;
    if (gy < 0 || gy >= IMG_W || gx < 0 || gx >= IMG_W) Tm[i] = 0.0f;
  }
  __syncthreads();

  const float inv9 = 1.0f / 9.0f;

  for (int chain = 0; chain < 2; ++chain) {
    // Stage 0 transform (both transforms map 0 -> 0, so OOB cells stay correctly zero).
    for (int i = tid; i < NLS; i += NTH) {
      float t = Tm[i];
      Ab[i] = (chain == 0) ? fmaxf(t - 0.75f, 0.0f) * 4.0f
                           : 1.0f - fmaxf(0.25f - t, 0.0f) * 4.0f;
    }
    __syncthreads();
    store_tile_channel(Ab, on, 3 + chain * 6, gy0, gx0, tid);

    for (int s = 1; s <= 5; ++s) {
      // Horizontal 3-tap sum: Ab -> Hb. Tile-edge contamination grows 1 col/stage and
      // after 5 stages reaches cols {0..4, 79..83}; the stored interior is cols 8..71.
      for (int i = tid; i < NLS; i += NTH) {
        int ty = i / LTW;
        int tx = i - ty * LTW;
        float c = Ab[i];
        float l = (tx > 0)       ? Ab[i - 1] : 0.0f;
        float r = (tx < LTW - 1) ? Ab[i + 1] : 0.0f;
        Hb[i] = l + c + r;
      }
      __syncthreads();
      // Vertical 3-tap sum * 1/9: Hb -> Ab, forcing image-OOB cells back to zero so the
      // next stage sees exact zero padding at image borders.
      for (int i = tid; i < NLS; i += NTH) {
        int ty = i / LTW;
        int tx = i - ty * LTW;
        float m = Hb[i];
        float u = (ty > 0)       ? Hb[i - LTW] : 0.0f;
        float d = (ty < LTH - 1) ? Hb[i + LTW] : 0.0f;
        int gy = gy0 + ty - HALO;
        int gx = gx0 + tx - XOFF;
        bool inb = (gy >= 0) && (gy < IMG_W) && (gx >= 0) && (gx < IMG_W);
        Ab[i] = inb ? (u + m + d) * inv9 : 0.0f;
      }
      __syncthreads();
      store_tile_channel(Ab, on, 3 + chain * 6 + s, gy0, gx0, tid);
    }
    __syncthreads();   // stores above read Ab; next chain's transform rewrites it
  }
}

extern "C" void kernel_launch(void* const* d_in, const int* in_sizes, int n_in,
                              void* d_out, int out_size, void* d_ws, size_t ws_size,
                              hipStream_t stream) {
  const float* x = (const float*)d_in[0];
  float* out = (float*)d_out;
  (void)d_ws; (void)ws_size; (void)n_in; (void)out_size;

  int batch = in_sizes[0] / (4 * IMG_W * IMG_W);   // = 8 for the reference shapes
  dim3 grid(IMG_W / TILE, IMG_W / TILE, batch);    // 16 x 16 x 8 = 2048 blocks
  matte_prep_kernel<<<grid, dim3(NTH, 1, 1), 0, stream>>>(x, out);
}